// ICP_7421703487935
// MI455X (gfx1250) — compile-verified
//
#include <hip/hip_runtime.h>
#include <math.h>

typedef float v2f __attribute__((ext_vector_type(2)));
typedef float v8f __attribute__((ext_vector_type(8)));

#define B_      4
#define N_      8192
#define M_      8192
#define TILES   (M_ / 16)      // 512 y-tiles of 16 points
#define ICP_ITERS 20
#define RMSE_THR  1e-5f

// ---------------- workspace layout (in floats) ----------------
// YB    : B * TILES * 32 * 2   = 131072   packed B-matrix tiles of Y
// XNN   : B * N * 3            =  98304   gathered nearest-neighbor coords
// PART  : B * 32 * 15          =   1920   per-block partial sums (means + cross)
// EPART : B * 32               =    128   per-block rmse partials
// RT    : B * 12               =     48   R (row-major) + T per batch
// PREV  : B                    =      4   previous rmse
// DONE  : 1 int
#define YB_OFF     0
#define XNN_OFF    131072
#define PART_OFF   229376
#define EPART_OFF  231296
#define RT_OFF     231424
#define PREV_OFF   231472
#define DONE_OFF   231476

// ---------------- init small state ----------------
__global__ void init_kernel(float* __restrict__ PREV, int* __restrict__ DONE) {
  int t = threadIdx.x;
  if (t < B_) PREV[t] = __builtin_huge_valf();
  if (t == 0) *DONE = 0;
}

// ---------------- pack Y into WMMA B-matrix tiles ----------------
// B (4x16) fragment layout assumed per ISA A-analog:
//   lanes 0-15 : VGPR0 = K0 row (y0), VGPR1 = K1 row (y1), N = lane
//   lanes16-31 : VGPR0 = K2 row (y2), VGPR1 = K3 row (|y|^2), N = lane-16
__global__ void ybuf_kernel(const float* __restrict__ Y, float* __restrict__ YB) {
  int b    = blockIdx.y;
  int lane = threadIdx.x & 31;
  int wid  = threadIdx.x >> 5;
  int t    = blockIdx.x * 8 + wid;            // tile index
  int m    = t * 16 + (lane & 15);            // y point index
  size_t base = ((size_t)b * M_ + m) * 3;
  float y0 = Y[base], y1 = Y[base + 1], y2 = Y[base + 2];
  float a0, a1;
  if (lane < 16) { a0 = y0; a1 = y1; }
  else           { a0 = y2; a1 = y0 * y0 + y1 * y1 + y2 * y2; }
  size_t o = (((size_t)b * TILES + t) * 32 + lane) * 2;
  YB[o] = a0; YB[o + 1] = a1;
}

// ---------------- NN search: WMMA 16x16x4 f32 ----------------
// Each wave handles 16 x-points; loops 512 y-tiles:
//   D[M][N] = |y_N|^2 - 2 x_M . y_N  (single wmma, C = 0)
// running min + argmin per (row, lane) with one v_cmp + two v_cndmask,
// then a 4-step shfl_xor butterfly within each 16-lane half, gather coords.
__global__ void nn_kernel(const float* __restrict__ X, const float* __restrict__ Y,
                          const float* __restrict__ YB, float* __restrict__ XNN) {
  int b    = blockIdx.y;
  int lane = threadIdx.x & 31;
  int wid  = threadIdx.x >> 5;
  int nbase = blockIdx.x * 128 + wid * 16;    // first of this wave's 16 x rows
  int m    = lane & 15;

  // A fragment (16x4): row M=m; lanes 0-15 hold K0,K1; lanes 16-31 hold K2,K3
  size_t xb = ((size_t)b * N_ + nbase + m) * 3;
  v2f a;
  if (lane < 16) { a.x = -2.0f * X[xb];     a.y = -2.0f * X[xb + 1]; }
  else           { a.x = -2.0f * X[xb + 2]; a.y = 1.0f; }

  float mv[8];
  int   mi[8];
#pragma unroll
  for (int r = 0; r < 8; ++r) { mv[r] = 3.4e38f; mi[r] = 0; }

  const v2f* yb = reinterpret_cast<const v2f*>(YB + (size_t)b * TILES * 64) + lane;

#pragma unroll 4
  for (int t = 0; t < TILES; ++t) {
    v2f bb = yb[t * 32];
    v8f c = {};
    v8f d = __builtin_amdgcn_wmma_f32_16x16x4_f32(
        /*neg_a=*/false, a, /*neg_b=*/false, bb,
        /*c_mod=*/(short)0, c, /*reuse_a=*/false, /*reuse_b=*/false);
    int cand = t * 16 + m;                    // column (y) index this lane holds
#pragma unroll
    for (int r = 0; r < 8; ++r) {
      float dv = d[r];
      bool lt  = dv < mv[r];                  // strict < keeps first (lowest t)
      mv[r] = lt ? dv   : mv[r];              // cndmask (shares VCC with idx)
      mi[r] = lt ? cand : mi[r];              // cndmask
    }
  }

  // per-row argmin across the 16 lanes holding that row
#pragma unroll
  for (int r = 0; r < 8; ++r) {
    float v = mv[r];
    int   i = mi[r];
#pragma unroll
    for (int off = 8; off >= 1; off >>= 1) {
      float ov = __shfl_xor(v, off, 16);
      int   oi = __shfl_xor(i, off, 16);
      bool take = (ov < v) || (ov == v && oi < i);   // tie -> lowest index
      v = take ? ov : v;
      i = take ? oi : i;
    }
    if (m == 0) {
      int row = r + ((lane >> 4) << 3);       // lanes16-31 hold rows 8..15
      size_t no = ((size_t)b * N_ + nbase + row) * 3;
      size_t yo = ((size_t)b * M_ + i) * 3;
      XNN[no]     = Y[yo];
      XNN[no + 1] = Y[yo + 1];
      XNN[no + 2] = Y[yo + 2];
    }
  }
}

// ---------------- stage-1 reduction: sums for means + cross-covariance ----------------
__global__ void reduce_kernel(const float* __restrict__ X, const float* __restrict__ T,
                              float* __restrict__ PART) {
  int b = blockIdx.y, tid = threadIdx.x;
  int n = blockIdx.x * 256 + tid;
  size_t base = ((size_t)b * N_ + n) * 3;
  float x0 = X[base], x1 = X[base + 1], x2 = X[base + 2];
  float y0 = T[base], y1 = T[base + 1], y2 = T[base + 2];
  float acc[15] = { x0, x1, x2, y0, y1, y2,
                    x0*y0, x0*y1, x0*y2,
                    x1*y0, x1*y1, x1*y2,
                    x2*y0, x2*y1, x2*y2 };
  __shared__ float sd[256];
  for (int c = 0; c < 15; ++c) {
    sd[tid] = acc[c];
    __syncthreads();
    for (int s = 128; s > 0; s >>= 1) {
      if (tid < s) sd[tid] += sd[tid + s];
      __syncthreads();
    }
    if (tid == 0) PART[((size_t)b * 32 + blockIdx.x) * 15 + c] = sd[0];
    __syncthreads();
  }
}

// ---------------- stage-2: Procrustes (3x3 Jacobi SVD + reflection fix) ----------------
__global__ void svd_kernel(const float* __restrict__ PART, float* __restrict__ RT) {
  int b = threadIdx.x;
  if (b >= B_) return;
  float S[15];
  for (int c = 0; c < 15; ++c) S[c] = 0.f;
  for (int blk = 0; blk < 32; ++blk) {
    const float* p = PART + ((size_t)b * 32 + blk) * 15;
    for (int c = 0; c < 15; ++c) S[c] += p[c];
  }
  const float invn = 1.0f / (float)N_;
  float xm[3], ym[3];
  for (int i = 0; i < 3; ++i) { xm[i] = S[i] * invn; ym[i] = S[3 + i] * invn; }
  float A[3][3];
  for (int i = 0; i < 3; ++i)
    for (int j = 0; j < 3; ++j)
      A[i][j] = S[6 + i * 3 + j] * invn - xm[i] * ym[j];   // cov = E[xy] - x̄ȳ

  // Jacobi eigendecomposition of Bm = A^T A -> V
  float Bm[3][3];
  for (int i = 0; i < 3; ++i)
    for (int j = 0; j < 3; ++j) {
      float s = 0.f;
      for (int k = 0; k < 3; ++k) s += A[k][i] * A[k][j];
      Bm[i][j] = s;
    }
  float V[3][3] = { {1,0,0},{0,1,0},{0,0,1} };
  for (int sweep = 0; sweep < 12; ++sweep) {
    for (int pair = 0; pair < 3; ++pair) {
      int p = (pair == 2) ? 1 : 0;
      int q = (pair == 0) ? 1 : 2;
      float apq = Bm[p][q];
      if (fabsf(apq) > 1e-20f) {
        float tau = (Bm[q][q] - Bm[p][p]) / (2.0f * apq);
        float tr  = (tau >= 0.f ? 1.f : -1.f) / (fabsf(tau) + sqrtf(1.0f + tau * tau));
        float c = 1.0f / sqrtf(1.0f + tr * tr);
        float s = tr * c;
        for (int k = 0; k < 3; ++k) { float bkp=Bm[k][p], bkq=Bm[k][q]; Bm[k][p]=c*bkp-s*bkq; Bm[k][q]=s*bkp+c*bkq; }
        for (int k = 0; k < 3; ++k) { float bpk=Bm[p][k], bqk=Bm[q][k]; Bm[p][k]=c*bpk-s*bqk; Bm[q][k]=s*bpk+c*bqk; }
        for (int k = 0; k < 3; ++k) { float vkp=V[k][p], vkq=V[k][q]; V[k][p]=c*vkp-s*vkq; V[k][q]=s*vkp+c*vkq; }
      }
    }
  }
  float lam[3] = { Bm[0][0], Bm[1][1], Bm[2][2] };
  for (int i = 0; i < 2; ++i)
    for (int j = i + 1; j < 3; ++j)
      if (lam[j] > lam[i]) {
        float tl = lam[i]; lam[i] = lam[j]; lam[j] = tl;
        for (int k = 0; k < 3; ++k) { float tv=V[k][i]; V[k][i]=V[k][j]; V[k][j]=tv; }
      }
  float sig0 = sqrtf(fmaxf(lam[0], 0.f));

  float U[3][3];
  for (int k = 0; k < 2; ++k) {
    float u0=0,u1=0,u2=0;
    for (int i = 0; i < 3; ++i) { u0+=A[0][i]*V[i][k]; u1+=A[1][i]*V[i][k]; u2+=A[2][i]*V[i][k]; }
    float nrm = sqrtf(u0*u0 + u1*u1 + u2*u2);
    float inv = (nrm > 1e-30f) ? 1.0f / nrm : 0.f;
    U[0][k]=u0*inv; U[1][k]=u1*inv; U[2][k]=u2*inv;
  }
  {
    float u0=0,u1=0,u2=0;
    for (int i = 0; i < 3; ++i) { u0+=A[0][i]*V[i][2]; u1+=A[1][i]*V[i][2]; u2+=A[2][i]*V[i][2]; }
    float nrm = sqrtf(u0*u0 + u1*u1 + u2*u2);
    if (nrm > 1e-6f * fmaxf(sig0, 1e-30f)) {
      float inv = 1.0f / nrm;
      U[0][2]=u0*inv; U[1][2]=u1*inv; U[2][2]=u2*inv;
    } else {   // degenerate: complete with cross(u0, u1)
      U[0][2] = U[1][0]*U[2][1] - U[2][0]*U[1][1];
      U[1][2] = U[2][0]*U[0][1] - U[0][0]*U[2][1];
      U[2][2] = U[0][0]*U[1][1] - U[1][0]*U[0][1];
    }
  }
  float detU = U[0][0]*(U[1][1]*U[2][2]-U[1][2]*U[2][1])
             - U[0][1]*(U[1][0]*U[2][2]-U[1][2]*U[2][0])
             + U[0][2]*(U[1][0]*U[2][1]-U[1][1]*U[2][0]);
  float detV = V[0][0]*(V[1][1]*V[2][2]-V[1][2]*V[2][1])
             - V[0][1]*(V[1][0]*V[2][2]-V[1][2]*V[2][0])
             + V[0][2]*(V[1][0]*V[2][1]-V[1][1]*V[2][0]);
  float e2 = (detU * detV < 0.f) ? -1.f : 1.f;

  float R[3][3];
  for (int i = 0; i < 3; ++i)
    for (int j = 0; j < 3; ++j)
      R[i][j] = U[i][0]*V[j][0] + U[i][1]*V[j][1] + e2 * U[i][2]*V[j][2];
  float Tt[3];
  for (int j = 0; j < 3; ++j)
    Tt[j] = ym[j] - (xm[0]*R[0][j] + xm[1]*R[1][j] + xm[2]*R[2][j]);

  float* out = RT + (size_t)b * 12;
  for (int i = 0; i < 3; ++i)
    for (int j = 0; j < 3; ++j) out[i * 3 + j] = R[i][j];
  for (int j = 0; j < 3; ++j) out[9 + j] = Tt[j];
}

// ---------------- apply transform + rmse partials ----------------
__global__ void apply_kernel(const float* __restrict__ Xsrc, const float* __restrict__ Tgt,
                             float* __restrict__ Xdst, const float* __restrict__ RT,
                             float* __restrict__ EPART, const int* __restrict__ DONE,
                             int is_init) {
  int b = blockIdx.y, tid = threadIdx.x;
  int n = blockIdx.x * 256 + tid;
  const float* rt = RT + (size_t)b * 12;
  float R00=rt[0],R01=rt[1],R02=rt[2],R10=rt[3],R11=rt[4],R12=rt[5],R20=rt[6],R21=rt[7],R22=rt[8];
  float T0=rt[9],T1=rt[10],T2=rt[11];
  size_t base = ((size_t)b * N_ + n) * 3;
  float x0 = Xsrc[base], x1 = Xsrc[base + 1], x2 = Xsrc[base + 2];
  float n0 = x0*R00 + x1*R10 + x2*R20 + T0;
  float n1 = x0*R01 + x1*R11 + x2*R21 + T1;
  float n2 = x0*R02 + x1*R12 + x2*R22 + T2;
  float t0 = Tgt[base], t1 = Tgt[base + 1], t2 = Tgt[base + 2];
  float d0 = n0 - t0, d1 = n1 - t1, d2 = n2 - t2;
  float err = d0*d0 + d1*d1 + d2*d2;
  int done = is_init ? 0 : *DONE;       // uses done from BEFORE this iter's update
  if (!done) { Xdst[base] = n0; Xdst[base + 1] = n1; Xdst[base + 2] = n2; }
  __shared__ float sd[256];
  sd[tid] = err;
  __syncthreads();
  for (int s = 128; s > 0; s >>= 1) {
    if (tid < s) sd[tid] += sd[tid + s];
    __syncthreads();
  }
  if (tid == 0) EPART[(size_t)b * 32 + blockIdx.x] = sd[0];
}

// ---------------- convergence flag ----------------
__global__ void flag_kernel(const float* __restrict__ EPART, float* __restrict__ PREV,
                            int* __restrict__ DONE) {
  __shared__ float relsh[B_];
  int t = threadIdx.x;
  int old = *DONE;
  if (t < B_) {
    float s = 0.f;
    for (int k = 0; k < 32; ++k) s += EPART[(size_t)t * 32 + k];
    float rmse = sqrtf(s / (float)N_);
    float pv = PREV[t];
    float rel = (pv <= 3.0e38f) ? fabsf(pv - rmse) / fmaxf(pv, 1e-30f) : 3.4e38f;
    relsh[t] = rel;
    if (!old) PREV[t] = rmse;           // rmse_out = where(done, prev, rmse)
  }
  __syncthreads();
  if (t == 0) {
    bool all = true;
    for (int k = 0; k < B_; ++k) all = all && (relsh[k] <= RMSE_THR);
    if (all) *DONE = 1;                 // new_done = done | all(rel<=thr)
  }
}

// ---------------- host orchestration ----------------
extern "C" void kernel_launch(void* const* d_in, const int* in_sizes, int n_in,
                              void* d_out, int out_size, void* d_ws, size_t ws_size,
                              hipStream_t stream) {
  (void)in_sizes; (void)n_in; (void)out_size; (void)ws_size;
  const float* Xin = (const float*)d_in[0];   // from_vertices [4,8192,3]
  const float* Yin = (const float*)d_in[1];   // to_vertices   [4,8192,3]
  float* Xt = (float*)d_out;                  // Xt lives in d_out throughout

  float* w     = (float*)d_ws;
  float* YB    = w + YB_OFF;
  float* XNN   = w + XNN_OFF;
  float* PART  = w + PART_OFF;
  float* EPART = w + EPART_OFF;
  float* RT    = w + RT_OFF;
  float* PREV  = w + PREV_OFF;
  int*   DONE  = (int*)(w + DONE_OFF);

  dim3 grid_red(32, B_), grid_nn(64, B_), grid_yb(64, B_);

  init_kernel<<<1, 32, 0, stream>>>(PREV, DONE);
  ybuf_kernel<<<grid_yb, 256, 0, stream>>>(Yin, YB);   // Y is fixed: pack once

  // initial paired Procrustes: align(X, Y), X0 = X@R + T  -> d_out
  reduce_kernel<<<grid_red, 256, 0, stream>>>(Xin, Yin, PART);
  svd_kernel<<<1, 32, 0, stream>>>(PART, RT);
  apply_kernel<<<grid_red, 256, 0, stream>>>(Xin, Yin, Xt, RT, EPART, DONE, /*is_init=*/1);

  for (int it = 0; it < ICP_ITERS; ++it) {
    nn_kernel<<<grid_nn, 256, 0, stream>>>(Xt, Yin, YB, XNN);
    reduce_kernel<<<grid_red, 256, 0, stream>>>(Xt, XNN, PART);
    svd_kernel<<<1, 32, 0, stream>>>(PART, RT);
    apply_kernel<<<grid_red, 256, 0, stream>>>(Xt, XNN, Xt, RT, EPART, DONE, /*is_init=*/0);
    flag_kernel<<<1, 32, 0, stream>>>(EPART, PREV, DONE);
  }
}